// ARIMAX_16432544875252
// MI455X (gfx1250) — compile-verified
//
#include <hip/hip_runtime.h>

#define AR_ORDER 8
#define EXO_DIM  8
#define SEQ_LEN  4096
#define BATCH    16384
#define UCOLS    (SEQ_LEN + EXO_DIM)       // 4104
#define OCOLS    (SEQ_LEN + AR_ORDER + 1)  // 4105

typedef __attribute__((ext_vector_type(2))) float v2f;
typedef __attribute__((ext_vector_type(8))) float v8f;

// ---------------------------------------------------------------------------
// Setup: build the 16x32 block-recurrence matrix W from w (device-resident).
//   x_{i+9} = sum_k a[k] x_{i+k} + g_i,   g_i = sum_j w_ex[j] u[i+1+j]
//   a[0] = -w_ar[0]; a[k] = w_ar[k-1]-w_ar[k] (k=1..7); a[8] = 1 + w_ar[7]
// Over a block of 16 steps: OUT[t] = sum_{k<9} W[t][k] x_{i+k}
//                                  + sum_{d<23} W[t][9+d] u[i+1+d]
// ---------------------------------------------------------------------------
__global__ void arimax_setup_W(const float* __restrict__ w, float* __restrict__ W)
{
    if (threadIdx.x != 0 || blockIdx.x != 0) return;
    float war[8], wex[8];
    for (int j = 0; j < 8; ++j) { war[j] = w[j]; wex[j] = w[8 + j]; }
    float a[9];
    a[0] = -war[0];
    for (int k = 1; k < 8; ++k) a[k] = war[k - 1] - war[k];
    a[8] = 1.0f + war[7];

    // c[m][*]: coefficients of x_{i+m} over basis [x_{i+0..8}, g_{i+0..15}]
    float c[25][25];
    for (int m = 0; m < 25; ++m)
        for (int j = 0; j < 25; ++j) c[m][j] = 0.0f;
    for (int m = 0; m < 9; ++m) c[m][m] = 1.0f;
    for (int m = 9; m < 25; ++m) {
        for (int k = 0; k < 9; ++k) {
            float ak = a[k];
            for (int j = 0; j < 25; ++j) c[m][j] += ak * c[m - 9 + k][j];
        }
        c[m][9 + (m - 9)] += 1.0f;   // + g_{m-9}
    }
    for (int t = 0; t < 16; ++t) {
        for (int k = 0; k < 9; ++k) W[t * 32 + k] = c[9 + t][k];
        for (int d = 0; d < 23; ++d) {           // fold FIR into Toeplitz cols
            float s = 0.0f;
            for (int q = 0; q < 16; ++q) {
                int j = d - q;
                if (j >= 0 && j < 8) s += c[9 + t][9 + q] * wex[j];
            }
            W[t * 32 + 9 + d] = s;
        }
    }
}

// ---------------------------------------------------------------------------
// Main: one wave32 per 16 batch rows. 256 blocks of 16 timesteps each; every
// block is OUT(16x16) = W(16x32) x X(32x16) via 8x V_WMMA_F32_16X16X4_F32
// (two independent accumulation chains). State for the next block = OUT rows
// 7..15; 4 upper->lower half shuffles (ds_bpermute), rest stays in-lane.
//
// Fully convergent hot loop: the lo/hi u-pair offsets unify to a single
// per-lane pointer urow + blk*16 + (hi?2:0) with six stride-4 pairs; the one
// junk element (lo d=-1) lands in the K=8 state slot and is cndmask'd away.
//
// Layouts (ISA 7.12.2, f32):
//   A 16x4 : lanes 0-15 hold M=lane, VGPR0=K0 VGPR1=K1; lanes 16-31: K2,K3
//   B 4x16 : lanes 0-15 hold N=lane, VGPR0=K0 VGPR1=K1; lanes 16-31: K2,K3
//   C/D    : VGPR v = (M=v, N=lane) for lanes 0-15, (M=v+8, N=lane-16) hi
// ---------------------------------------------------------------------------
__global__ __launch_bounds__(256) void arimax_main(
    const float* __restrict__ y, const float* __restrict__ u,
    const float* __restrict__ W, float* __restrict__ out)
{
    const int lane = threadIdx.x & 31;
    const int wave = blockIdx.x * (blockDim.x >> 5) + (threadIdx.x >> 5);
    const int r16  = lane & 15;
    const bool hi  = lane >= 16;
    const int row  = wave * 16 + r16;

    const float* __restrict__ urow = u + (size_t)row * UCOLS;
    const float* __restrict__ yrow = y + (size_t)row * (AR_ORDER + 1);
    float* __restrict__ orow       = out + (size_t)row * OCOLS;

    // A operand: W rows striped per the 16x4 f32 A layout (per-K-slice kk).
    v2f a[8];
    {
        const float* wr = W + r16 * 32 + (hi ? 2 : 0);
        #pragma unroll
        for (int kk = 0; kk < 8; ++kk) { a[kk].x = wr[4 * kk]; a[kk].y = wr[4 * kk + 1]; }
    }

    // Per-lane u pointer: lo lanes read pairs at d = 4k-1, hi at d = 4k+1.
    const float* __restrict__ ulane = urow + (hi ? 2 : 0);

    // Initial state (K=0..8) from y, composed branchlessly.
    v2f b0, b1;
    b0.x = hi ? yrow[2] : yrow[0];        // K=2 / K=0
    b0.y = hi ? yrow[3] : yrow[1];        // K=3 / K=1
    b1.x = hi ? yrow[6] : yrow[4];        // K=6 / K=4
    b1.y = hi ? yrow[7] : yrow[5];        // K=7 / K=5
    float s15c = yrow[8];                 // K=8 carry (lo lanes only use it)

    if (!hi) {                            // out[:, :9] = y (once, tiny)
        #pragma unroll
        for (int k = 0; k < 9; ++k) orow[k] = yrow[k];
    }

    for (int blk = 0; blk < SEQ_LEN / 16; ++blk) {
        const float* up = ulane + blk * 16;

        // Six full-wave stride-4 pairs -> b2..b7 (compiler merges to b64).
        v2f b2, b3, b4, b5, b6, b7;
        float u0 = up[0];
        b2.y = up[1];
        b3.x = up[4];  b3.y = up[5];
        b4.x = up[8];  b4.y = up[9];
        b5.x = up[12]; b5.y = up[13];
        b6.x = up[16]; b6.y = up[17];
        b7.x = up[20]; b7.y = up[21];
        b2.x = hi ? u0 : s15c;            // K=10 (u) / K=8 (state)
        __builtin_prefetch(up + 24, 0, 1);  // next block's fresh lines -> GL2

        // OUT = W x X : 8 K-slices, two independent accumulation chains.
        v8f c0 = {}, c1 = {};
        c0 = __builtin_amdgcn_wmma_f32_16x16x4_f32(false, a[0], false, b0, (short)0, c0, false, false);
        c1 = __builtin_amdgcn_wmma_f32_16x16x4_f32(false, a[1], false, b1, (short)0, c1, false, false);
        c0 = __builtin_amdgcn_wmma_f32_16x16x4_f32(false, a[2], false, b2, (short)0, c0, false, false);
        c1 = __builtin_amdgcn_wmma_f32_16x16x4_f32(false, a[3], false, b3, (short)0, c1, false, false);
        c0 = __builtin_amdgcn_wmma_f32_16x16x4_f32(false, a[4], false, b4, (short)0, c0, false, false);
        c1 = __builtin_amdgcn_wmma_f32_16x16x4_f32(false, a[5], false, b5, (short)0, c1, false, false);
        c0 = __builtin_amdgcn_wmma_f32_16x16x4_f32(false, a[6], false, b6, (short)0, c0, false, false);
        c1 = __builtin_amdgcn_wmma_f32_16x16x4_f32(false, a[7], false, b7, (short)0, c1, false, false);

        v8f d;
        #pragma unroll
        for (int v = 0; v < 8; ++v) d[v] = c0[v] + c1[v];

        // Store: lower lanes hold x_{i+9..i+16}, upper lanes x_{i+17..i+24}
        // (8 contiguous floats per lane -> merged b128 stores).
        const int tb = blk * 16 + 9 + (hi ? 8 : 0);
        #pragma unroll
        for (int v = 0; v < 8; ++v) orow[tb + v] = d[v];

        // Next state = OUT rows 7..15. Upper->lower half moves for K=1,4,5,8.
        const int p = r16 + 16;
        float s8  = __shfl(d[0], p, 32);   // x_{i+17} (t=8)
        float s11 = __shfl(d[3], p, 32);   // x_{i+20} (t=11)
        float s12 = __shfl(d[4], p, 32);   // x_{i+21} (t=12)
        s15c      = __shfl(d[7], p, 32);   // x_{i+24} (t=15) -> next b2.x (lo)
        b0.x = hi ? d[1] : d[7];           // K=2: x_{i+18} / K=0: x_{i+16}
        b0.y = hi ? d[2] : s8;             // K=3: x_{i+19} / K=1: x_{i+17}
        b1.x = hi ? d[5] : s11;            // K=6: x_{i+22} / K=4: x_{i+20}
        b1.y = hi ? d[6] : s12;            // K=7: x_{i+23} / K=5: x_{i+21}
    }
}

extern "C" void kernel_launch(void* const* d_in, const int* in_sizes, int n_in,
                              void* d_out, int out_size, void* d_ws, size_t ws_size,
                              hipStream_t stream)
{
    const float* y = (const float*)d_in[0];
    const float* u = (const float*)d_in[1];
    const float* w = (const float*)d_in[2];
    float* out = (float*)d_out;
    float* Wmat = (float*)d_ws;            // 16*32 floats = 2 KB scratch

    arimax_setup_W<<<1, 1, 0, stream>>>(w, Wmat);

    const int waves = BATCH / 16;          // 1024 waves, 16 rows each
    const int wavesPerBlock = 8;           // 256 threads
    arimax_main<<<waves / wavesPerBlock, wavesPerBlock * 32, 0, stream>>>(y, u, Wmat, out);
}